// GCNLinkPrediction_20529943675093
// MI455X (gfx1250) — compile-verified
//
#include <hip/hip_runtime.h>
#include <hip/hip_bf16.h>

// GCN link prediction for MI455X (gfx1250, wave32).
// Pipeline: deg -> rsqrt -> [WMMA f32 GEMM + fused bias/self-loop] -> edge
// scatter (fp32 global atomics) -> ReLU -> GEMM2 -> scatter -> pair-dot decode.

typedef float v2f __attribute__((ext_vector_type(2)));
typedef float v8f __attribute__((ext_vector_type(8)));

#define H_DIM 64

// ---------------- degree / norm ----------------
__global__ void k_deg_init(float* __restrict__ deg, int n) {
  int i = blockIdx.x * blockDim.x + threadIdx.x;
  if (i < n) deg[i] = 1.0f;  // self-loop contributes 1 to every node's degree
}

__global__ void k_deg_accum(const int* __restrict__ dst, float* __restrict__ deg,
                            int ne) {
  int i = blockIdx.x * blockDim.x + threadIdx.x;
  if (i < ne) unsafeAtomicAdd(&deg[dst[i]], 1.0f);
}

__global__ void k_deg_rsqrt(float* __restrict__ deg, int n) {
  int i = blockIdx.x * blockDim.x + threadIdx.x;
  if (i < n) deg[i] = rsqrtf(deg[i]);  // deg >= 1 always (self-loops)
}

// ---------------- dense transform: H = X @ W, Acc = bias + H*dis^2 ----------
// One block = 128 threads = 4 waves; block handles 16 rows, wave w handles
// columns [16w, 16w+16). 16 WMMA f32 16x16x4 steps cover K=64.
__global__ void k_gemm_wmma_fused(const float* __restrict__ X,
                                  const float* __restrict__ W,
                                  const float* __restrict__ bias,
                                  const float* __restrict__ dis,
                                  float* __restrict__ Hout,
                                  float* __restrict__ Acc, int n) {
  const int lane  = threadIdx.x & 31;
  const int wave  = threadIdx.x >> 5;     // 0..3 -> 16-col chunk of H
  const int row0  = blockIdx.x * 16;
  const int n0    = wave * 16;
  const int m     = lane & 15;
  const int khalf = (lane >> 4) << 1;     // lanes 0-15: K offset 0; 16-31: 2

  int arow = row0 + m;
  if (arow >= n) arow = n - 1;            // clamp for tail tiles (reads only)
  const float* __restrict__ xrow = X + (size_t)arow * H_DIM;

  v8f acc = {};
#pragma unroll
  for (int kk = 0; kk < 16; ++kk) {
    const int kb = kk * 4 + khalf;
    v2f a;                                 // A 16x4: lane -> rows, K pair
    a.x = xrow[kb];
    a.y = xrow[kb + 1];
    v2f b;                                 // B 4x16: lane -> cols, K pair
    b.x = W[kb * H_DIM + n0 + m];
    b.y = W[(kb + 1) * H_DIM + n0 + m];
    acc = __builtin_amdgcn_wmma_f32_16x16x4_f32(false, a, false, b, (short)0,
                                                acc, false, false);
  }

  // C/D layout: VGPR r -> row r + 8*(lane>=16), col = lane&15 (+ n0)
  const int rbase = (lane >> 4) << 3;
  const int col   = n0 + m;
  const float bc  = bias[col];
#pragma unroll
  for (int r = 0; r < 8; ++r) {
    const int row = row0 + rbase + r;
    if (row < n) {
      const float h = acc[r];
      const float d = dis[row];
      Hout[(size_t)row * H_DIM + col] = h;
      Acc[(size_t)row * H_DIM + col] = fmaf(h, d * d, bc);  // bias + self-loop
    }
  }
}

// ---------------- edge scatter: Acc[dst] += H[src] * dis[src]*dis[dst] ------
// One wave32 per edge; each lane owns a float2 column pair (64 cols).
__global__ void k_scatter_edges(const float* __restrict__ Hm,
                                const int* __restrict__ src,
                                const int* __restrict__ dst,
                                const float* __restrict__ dis,
                                float* __restrict__ Acc, int ne) {
  const int e = blockIdx.x * (blockDim.x >> 5) + (threadIdx.x >> 5);
  if (e >= ne) return;
  const int lane = threadIdx.x & 31;
  const int s = src[e];                    // wave-uniform -> scalar loads
  const int d = dst[e];
  const float nrm = dis[s] * dis[d];
  const float2 v = ((const float2*)(Hm + (size_t)s * H_DIM))[lane];
  float* outp = Acc + (size_t)d * H_DIM + lane * 2;
  unsafeAtomicAdd(outp, v.x * nrm);
  unsafeAtomicAdd(outp + 1, v.y * nrm);
}

// ---------------- ReLU ----------------
__global__ void k_relu(float* __restrict__ a, size_t n) {
  size_t i = (size_t)blockIdx.x * blockDim.x + threadIdx.x;
  if (i < n) a[i] = fmaxf(a[i], 0.0f);
}

// ---------------- decode: logits[k] = dot(Z[a_k], Z[b_k]) -------------------
__global__ void k_decode(const float* __restrict__ Z,
                         const int* __restrict__ pos,
                         const int* __restrict__ neg,
                         float* __restrict__ out, int npos, int nneg) {
  const int k = blockIdx.x * (blockDim.x >> 5) + (threadIdx.x >> 5);
  if (k >= npos + nneg) return;
  const int lane = threadIdx.x & 31;
  int a, b;
  if (k < npos) {
    a = pos[k];
    b = pos[npos + k];
  } else {
    const int j = k - npos;
    a = neg[j];
    b = neg[nneg + j];
  }
  const float2 va = ((const float2*)(Z + (size_t)a * H_DIM))[lane];
  const float2 vb = ((const float2*)(Z + (size_t)b * H_DIM))[lane];
  float s = fmaf(va.x, vb.x, va.y * vb.y);
#pragma unroll
  for (int off = 16; off > 0; off >>= 1) s += __shfl_xor(s, off, 32);
  if (lane == 0) out[k] = s;
}

// ---------------- host launcher ----------------
extern "C" void kernel_launch(void* const* d_in, const int* in_sizes, int n_in,
                              void* d_out, int out_size, void* d_ws,
                              size_t ws_size, hipStream_t stream) {
  (void)n_in; (void)out_size; (void)ws_size;
  const float* x   = (const float*)d_in[0];
  const int*   ei  = (const int*)d_in[1];   // [2, E] row-major
  const int*   pei = (const int*)d_in[2];   // [2, PE]
  const int*   nei = (const int*)d_in[3];   // [2, NE]
  const float* W1  = (const float*)d_in[4]; // [64, 64]
  const float* b1  = (const float*)d_in[5];
  const float* W2  = (const float*)d_in[6];
  const float* b2  = (const float*)d_in[7];
  float* out = (float*)d_out;

  const int n    = in_sizes[0] / H_DIM;
  const int ne   = in_sizes[1] / 2;
  const int npos = in_sizes[2] / 2;
  const int nneg = in_sizes[3] / 2;
  const int* src = ei;
  const int* dst = ei + ne;

  // Workspace layout (floats): dis[N] | h[N*64] | acc1[N*64] | acc2[N*64]
  float* ws = (float*)d_ws;
  const size_t nAl = ((size_t)n + 63) & ~(size_t)63;
  float* dis  = ws;
  float* hbuf = ws + nAl;
  float* acc1 = hbuf + (size_t)n * H_DIM;
  float* acc2 = acc1 + (size_t)n * H_DIM;

  const int B = 256;
  k_deg_init<<<(n + B - 1) / B, B, 0, stream>>>(dis, n);
  k_deg_accum<<<(ne + B - 1) / B, B, 0, stream>>>(dst, dis, ne);
  k_deg_rsqrt<<<(n + B - 1) / B, B, 0, stream>>>(dis, n);

  const int tiles = (n + 15) / 16;
  // Layer 1
  k_gemm_wmma_fused<<<tiles, 128, 0, stream>>>(x, W1, b1, dis, hbuf, acc1, n);
  k_scatter_edges<<<(ne + 7) / 8, B, 0, stream>>>(hbuf, src, dst, dis, acc1, ne);
  k_relu<<<(int)(((size_t)n * H_DIM + B - 1) / B), B, 0, stream>>>(
      acc1, (size_t)n * H_DIM);
  // Layer 2
  k_gemm_wmma_fused<<<tiles, 128, 0, stream>>>(acc1, W2, b2, dis, hbuf, acc2, n);
  k_scatter_edges<<<(ne + 7) / 8, B, 0, stream>>>(hbuf, src, dst, dis, acc2, ne);
  // Decode
  k_decode<<<(npos + nneg + 7) / 8, B, 0, stream>>>(acc2, pei, nei, out, npos,
                                                    nneg);
}